// PolarIoULoss_37649683317009
// MI455X (gfx1250) — compile-verified
//
#include <hip/hip_runtime.h>
#include <math.h>

typedef __attribute__((ext_vector_type(2))) float v2f;
typedef __attribute__((ext_vector_type(8))) float v8f;

#define D_COLS 36
#define BLOCK_THREADS 256
#define WAVES_PER_BLOCK (BLOCK_THREADS / 32)
#define GRID_BLOCKS 2048

// Kernel 1: each wave processes groups of 16 rows.
// Load 16x4 fp32 tiles of the inputs in WMMA A-layout (float2 per lane: lanes
// 0-15 hold K=0,1 / lanes 16-31 hold K=2,3 of row lane&15), compute max/min
// elementwise, and reduce rows with V_WMMA_F32_16X16X4_F32 against an all-ones
// B matrix -> exact fp32 row sums in the C tile. All 18 loads for a group are
// issued before the compute chain to maximize outstanding loads per wave.
__global__ __launch_bounds__(BLOCK_THREADS) void polar_iou_partial_kernel(
    const float* __restrict__ yp, const float* __restrict__ yg,
    float* __restrict__ partial, unsigned int numGroups)
{
    const int lane = threadIdx.x & 31;
    const int wave = threadIdx.x >> 5;
    const unsigned int waveGlobal = blockIdx.x * WAVES_PER_BLOCK + wave;
    const unsigned int totalWaves = gridDim.x * WAVES_PER_BLOCK;

    const int rowInGroup = lane & 15;
    const int colBase    = (lane >> 4) << 1;   // 0 or 2

    const v2f onesB = {1.0f, 1.0f};            // B = all ones -> layout-invariant
    float acc = 0.0f;

    for (unsigned int g = waveGlobal; g < numGroups; g += totalWaves) {
        const unsigned long long row = (unsigned long long)g * 16ull + (unsigned)rowInGroup;
        const float* prow = yp + row * D_COLS + colBase;
        const float* grow = yg + row * D_COLS + colBase;

        // Issue all 18 b64 loads for this group up front (distinct registers)
        // so the wave keeps many loads in flight instead of one wait per step.
        v2f pd[9], gd[9];
#pragma unroll
        for (int s = 0; s < 9; ++s) pd[s] = *(const v2f*)(prow + 4 * s);
#pragma unroll
        for (int s = 0; s < 9; ++s) gd[s] = *(const v2f*)(grow + 4 * s);

        // Prefetch the next grid-stride tile (speculative, OOB-safe).
        {
            const unsigned long long nrow = row + (unsigned long long)totalWaves * 16ull;
            __builtin_prefetch(yp + nrow * D_COLS + colBase, 0, 1);
            __builtin_prefetch(yg + nrow * D_COLS + colBase, 0, 1);
        }

        v8f cmax = {};
        v8f cmin = {};
#pragma unroll
        for (int s = 0; s < 9; ++s) {          // 9 K-chunks of 4 cover D=36
            v2f p = pd[s];
            v2f q = gd[s];
            v2f amax = { fmaxf(p.x, q.x), fmaxf(p.y, q.y) };
            v2f amin = { fminf(p.x, q.x), fminf(p.y, q.y) };
            cmax = __builtin_amdgcn_wmma_f32_16x16x4_f32(
                false, amax, false, onesB, (short)0, cmax, false, false);
            cmin = __builtin_amdgcn_wmma_f32_16x16x4_f32(
                false, amin, false, onesB, (short)0, cmin, false, false);
        }

        // C/D layout: VGPR v, lanes 0-15 -> row v; lanes 16-31 -> row v+8.
        // Each lane selects its VGPR (lane&7); rows 8..15 fetched from lanes 16..23.
        const int v = lane & 7;
        float smax =
            v == 0 ? cmax[0] : v == 1 ? cmax[1] : v == 2 ? cmax[2] : v == 3 ? cmax[3] :
            v == 4 ? cmax[4] : v == 5 ? cmax[5] : v == 6 ? cmax[6] : cmax[7];
        float smin =
            v == 0 ? cmin[0] : v == 1 ? cmin[1] : v == 2 ? cmin[2] : v == 3 ? cmin[3] :
            v == 4 ? cmin[4] : v == 5 ? cmin[5] : v == 6 ? cmin[6] : cmin[7];

        const int src = (lane < 8) ? lane : lane + 8;  // lanes >=16: don't-care
        float rmax = __shfl(smax, src, 32);
        float rmin = __shfl(smin, src, 32);

        float lg = logf(rmax / rmin);          // all lanes hold valid positive sums
        acc += (lane < 16) ? lg : 0.0f;        // mask duplicate rows
    }

    // Wave reduction (wave32).
#pragma unroll
    for (int off = 16; off > 0; off >>= 1)
        acc += __shfl_xor(acc, off, 32);

    __shared__ float wsum[WAVES_PER_BLOCK];
    if (lane == 0) wsum[wave] = acc;
    __syncthreads();
    if (threadIdx.x == 0) {
        float s = 0.0f;
#pragma unroll
        for (int w = 0; w < WAVES_PER_BLOCK; ++w) s += wsum[w];
        partial[blockIdx.x] = s;               // each block owns exactly one slot
    }
}

// Kernel 2: deterministic final reduction + generic tail rows + mean.
__global__ __launch_bounds__(BLOCK_THREADS) void polar_iou_final_kernel(
    const float* __restrict__ partial, int nPartials,
    const float* __restrict__ yp, const float* __restrict__ yg,
    unsigned long long tailStart, unsigned long long Brows,
    float* __restrict__ out)
{
    __shared__ float sdata[BLOCK_THREADS];
    float s = 0.0f;
    for (int i = threadIdx.x; i < nPartials; i += BLOCK_THREADS) s += partial[i];

    // Tail rows not covered by 16-row groups (none for B=524288; kept for generality).
    for (unsigned long long r = tailStart + threadIdx.x; r < Brows; r += BLOCK_THREADS) {
        float smax = 0.0f, smin = 0.0f;
        for (int j = 0; j < D_COLS; ++j) {
            float p = yp[r * D_COLS + j];
            float q = yg[r * D_COLS + j];
            smax += fmaxf(p, q);
            smin += fminf(p, q);
        }
        s += logf(smax / smin);
    }

    sdata[threadIdx.x] = s;
    __syncthreads();
    for (int st = BLOCK_THREADS / 2; st > 0; st >>= 1) {
        if ((int)threadIdx.x < st) sdata[threadIdx.x] += sdata[threadIdx.x + st];
        __syncthreads();
    }
    if (threadIdx.x == 0) out[0] = sdata[0] / (float)Brows;
}

extern "C" void kernel_launch(void* const* d_in, const int* in_sizes, int n_in,
                              void* d_out, int out_size, void* d_ws, size_t ws_size,
                              hipStream_t stream) {
    const float* yp = (const float*)d_in[0];
    const float* yg = (const float*)d_in[1];
    float* out = (float*)d_out;
    float* partial = (float*)d_ws;             // GRID_BLOCKS floats (8 KB)

    const unsigned long long Brows = (unsigned long long)in_sizes[0] / D_COLS;
    const unsigned int numGroups = (unsigned int)(Brows / 16ull);

    polar_iou_partial_kernel<<<GRID_BLOCKS, BLOCK_THREADS, 0, stream>>>(
        yp, yg, partial, numGroups);
    polar_iou_final_kernel<<<1, BLOCK_THREADS, 0, stream>>>(
        partial, GRID_BLOCKS, yp, yg, (unsigned long long)numGroups * 16ull, Brows, out);
}